// mrnLoss_5119601017566
// MI455X (gfx1250) — compile-verified
//
#include <hip/hip_runtime.h>

typedef __attribute__((ext_vector_type(2))) float v2f;
typedef __attribute__((ext_vector_type(8))) float v8f;

#define LAM 1.0f

// ---------------------------------------------------------------------------
// out[0] = lam * N   (the "+1 per sample" part of perf_loss), overwriting the
// poisoned output buffer. Everything else atomically accumulates on top.
// ---------------------------------------------------------------------------
__global__ void mrn_init_kernel(float* __restrict__ out, float base) {
    if (threadIdx.x == 0 && blockIdx.x == 0) out[0] = base;
}

// ---------------------------------------------------------------------------
// reg_loss = sum((w_preds - w_stars)^2) over C*D elements (4 MB total; cheap)
// ---------------------------------------------------------------------------
__global__ void mrn_reg_kernel(const float* __restrict__ wp,
                               const float* __restrict__ ws,
                               float* __restrict__ out, int n) {
    float s = 0.0f;
    for (int i = blockIdx.x * blockDim.x + threadIdx.x; i < n;
         i += gridDim.x * blockDim.x) {
        float d = wp[i] - ws[i];
        s = fmaf(d, d, s);
    }
    // wave32 reduction
    #pragma unroll
    for (int off = 16; off > 0; off >>= 1) s += __shfl_xor(s, off, 32);

    __shared__ float bsum;
    if (threadIdx.x == 0) bsum = 0.0f;
    __syncthreads();
    if ((threadIdx.x & 31) == 0) atomicAdd(&bsum, s);
    __syncthreads();
    if (threadIdx.x == 0) atomicAdd(out, bsum);
}

// ---------------------------------------------------------------------------
// perf kernel: each wave computes 16 sample dots with V_WMMA_F32_16X16X4_F32.
//   A (16x4) : 16 samples' feats chunk   -> lane m (0-15): K=0,1 ; lane m+16: K=2,3
//   B (4x16) : gathered w_preds rows     -> same lane-symmetric layout
//   D (16x16): diagonal D[i][i] = dot(feats_i, w_preds[seg_i]) over this K-chunk
// Accumulate over D/4 = 128 chunks, read diagonal, fold y_n, reduce, atomic.
// Contribution to out is  -lam * sum(y_n * dot_n).
// ---------------------------------------------------------------------------
__global__ void mrn_perf_wmma_kernel(const float* __restrict__ w_preds,
                                     const float* __restrict__ feats,
                                     const float* __restrict__ ys,
                                     const int*   __restrict__ seg_ids,
                                     float* __restrict__ out,
                                     int N, int D) {
    const int lane = threadIdx.x & 31;
    const int waveInBlock = threadIdx.x >> 5;
    const long tile = (long)blockIdx.x * (blockDim.x >> 5) + waveInBlock;
    const long s0 = tile * 16;

    __shared__ float bsum;
    if (threadIdx.x == 0) bsum = 0.0f;
    __syncthreads();

    if (s0 + 16 <= (long)N) {   // wave-uniform branch: EXEC all-1s inside
        const int row  = lane & 15;          // sample within tile
        const int koff = (lane >> 4) << 1;   // 0 for lanes 0-15, 2 for 16-31
        const long sidx = s0 + row;

        const float* __restrict__ frow = feats   + sidx * (long)D;
        const float* __restrict__ wrow = w_preds + (long)seg_ids[sidx] * (long)D;

        v8f acc = {};
        #pragma unroll 4
        for (int k = 0; k < D; k += 4) {
            v2f a = __builtin_nontemporal_load((const v2f*)(frow + k + koff));
            v2f b = *(const v2f*)(wrow + k + koff);
            acc = __builtin_amdgcn_wmma_f32_16x16x4_f32(
                /*neg_a=*/false, a, /*neg_b=*/false, b,
                /*c_mod=*/(short)0, acc, /*reuse_a=*/false, /*reuse_b=*/false);
        }

        // Diagonal: lanes 0-7 hold D[i][i] in acc[lane] (M=i, N=i);
        //           lanes 24-31 hold D[i][i] (i=8..15) in acc[lane-24].
        const bool active = (lane < 8) || (lane >= 24);
        const int  rsel   = (lane < 8) ? lane : (lane - 24);
        float diag = 0.0f;
        #pragma unroll
        for (int j = 0; j < 8; ++j) diag = (rsel == j) ? acc[j] : diag;

        float term = 0.0f;
        if (active) {
            const long m = s0 + ((lane < 8) ? lane : (lane - 16));
            term = ys[m] * diag;
        }
        #pragma unroll
        for (int off = 16; off > 0; off >>= 1) term += __shfl_xor(term, off, 32);
        if (lane == 0) atomicAdd(&bsum, term);
    }

    __syncthreads();
    if (threadIdx.x == 0) atomicAdd(out, -LAM * bsum);
}

// ---------------------------------------------------------------------------
// Scalar tail for N % 16 samples (N=200000 -> tail 0, but keep it general).
// ---------------------------------------------------------------------------
__global__ void mrn_tail_kernel(const float* __restrict__ w_preds,
                                const float* __restrict__ feats,
                                const float* __restrict__ ys,
                                const int*   __restrict__ seg_ids,
                                float* __restrict__ out,
                                int start, int N, int D) {
    int i = start + blockIdx.x * blockDim.x + threadIdx.x;
    if (i >= N) return;
    const float* f = feats   + (long)i * D;
    const float* w = w_preds + (long)seg_ids[i] * D;
    float s = 0.0f;
    for (int k = 0; k < D; ++k) s = fmaf(f[k], w[k], s);
    atomicAdd(out, -LAM * ys[i] * s);
}

extern "C" void kernel_launch(void* const* d_in, const int* in_sizes, int n_in,
                              void* d_out, int out_size, void* d_ws, size_t ws_size,
                              hipStream_t stream) {
    const float* w_preds = (const float*)d_in[0];
    const float* w_stars = (const float*)d_in[1];
    const float* feats   = (const float*)d_in[2];
    const float* ys      = (const float*)d_in[3];
    const int*   seg_ids = (const int*)d_in[4];
    float* out = (float*)d_out;

    const int CD = in_sizes[0];            // C*D
    const int N  = in_sizes[3];            // number of samples
    const int D  = in_sizes[2] / N;        // feature dim (512)

    // 1) out = lam * N
    mrn_init_kernel<<<1, 1, 0, stream>>>(out, LAM * (float)N);

    // 2) += reg_loss
    {
        int threads = 256;
        int blocks = (CD + threads - 1) / threads;
        if (blocks > 1024) blocks = 1024;
        mrn_reg_kernel<<<blocks, threads, 0, stream>>>(w_preds, w_stars, out, CD);
    }

    // 3) -= lam * sum(y * dot) via WMMA (16 samples per wave, 8 waves/block)
    {
        long tiles = (long)N / 16;
        int  wavesPerBlock = 8;            // 256 threads
        long blocks = (tiles + wavesPerBlock - 1) / wavesPerBlock;
        if (blocks > 0)
            mrn_perf_wmma_kernel<<<(int)blocks, 256, 0, stream>>>(
                w_preds, feats, ys, seg_ids, out, N, D);
    }

    // 4) scalar tail (none for N=200000)
    {
        int tail = N % 16;
        if (tail > 0) {
            int start = N - tail;
            mrn_tail_kernel<<<1, 64, 0, stream>>>(
                w_preds, feats, ys, seg_ids, out, start, N, D);
        }
    }
}